// GCN_pyg_40785009443359
// MI455X (gfx1250) — compile-verified
//
#include <hip/hip_runtime.h>
#include <hip/hip_bf16.h>

#define NN 100000
#define EE 1600000
#define DD 128
#define GG 1024
#define LL 3
#define BN_EPS 1e-5f

typedef __attribute__((ext_vector_type(2))) float v2f;
typedef __attribute__((ext_vector_type(8))) float v8f;

// ---------------- small utility kernels ----------------

__global__ void fill_f32(float* p, float v, int n) {
    int i = blockIdx.x * blockDim.x + threadIdx.x;
    if (i < n) p[i] = v;
}

__global__ void deg_accum(const int* __restrict__ ei, float* __restrict__ deg) {
    int e = blockIdx.x * blockDim.x + threadIdx.x;
    if (e < EE) {
        int dst = ei[EE + e];
        unsafeAtomicAdd(&deg[dst], 1.0f);
    }
}

__global__ void deg_to_dinv(float* __restrict__ deg) {
    int i = blockIdx.x * blockDim.x + threadIdx.x;
    if (i < NN) deg[i] = __frsqrt_rn(deg[i]);
}

// ---------------- fused dual-GEMM (WMMA f32 16x16x4) ----------------
// Per block: 16 rows of x staged into LDS via CDNA5 async global->LDS DMA
// (global_load_async_to_lds_b128, ASYNCcnt). 8 waves each own one 16-col
// tile of the D=128 output. Each wave runs two WMMA chains sharing the A
// operand:
//   h   = x @ W          (stored to h_out for edge gather)
//   acc = relu(x@rW+rb) + b + h * dinv[row]^2   (self-loop folded in)
__global__ __launch_bounds__(256) void gemm_dual_wmma(
    const float* __restrict__ x,
    const float* __restrict__ W, const float* __restrict__ rW,
    const float* __restrict__ b, const float* __restrict__ rb,
    const float* __restrict__ dinv,
    float* __restrict__ h_out, float* __restrict__ acc_out)
{
    __shared__ __align__(16) float xs[16 * DD];  // 8 KB tile of x

    const int rowbase = blockIdx.x * 16;
    const int tid  = threadIdx.x;
    const int lane = tid & 31;
    const int wave = tid >> 5;                   // 0..7 -> column tile
    const int colbase = wave * 16;

    // async-stage 16x128 tile: each lane DMAs 16B directly into LDS,
    // 2 issues per thread cover the whole tile (512 x b128).
    {
        const unsigned lds_base = (unsigned)(size_t)&xs[0];
        #pragma unroll
        for (int it = 0; it < 2; ++it) {
            const int idx = tid + it * 256;      // 0..511
            const int r   = idx >> 5;            // / (DD/4)
            const int c4  = idx & 31;
            const float* g = x + (size_t)(rowbase + r) * DD + (c4 << 2);
            const unsigned l = lds_base + (unsigned)(idx * 16);
            asm volatile("global_load_async_to_lds_b128 %0, %1, off"
                         :: "v"(l), "v"(g) : "memory");
        }
        asm volatile("s_wait_asynccnt 0x0" ::: "memory");
    }
    __syncthreads();

    const int m    = lane & 15;                  // row (A) / col (B) within tile
    const int ksel = lane >> 4;                  // 0: K0,K1  1: K2,K3

    v8f hacc = {};
    v8f racc = {};

    #pragma unroll 4
    for (int k = 0; k < DD; k += 4) {
        const int k0 = k + 2 * ksel;
        v2f a, bw, br;
        a.x  = xs[m * DD + k0 + 0];
        a.y  = xs[m * DD + k0 + 1];
        bw.x = W [(size_t)(k0 + 0) * DD + colbase + m];
        bw.y = W [(size_t)(k0 + 1) * DD + colbase + m];
        br.x = rW[(size_t)(k0 + 0) * DD + colbase + m];
        br.y = rW[(size_t)(k0 + 1) * DD + colbase + m];
        hacc = __builtin_amdgcn_wmma_f32_16x16x4_f32(
                   false, a, false, bw, (short)0, hacc, false, false);
        racc = __builtin_amdgcn_wmma_f32_16x16x4_f32(
                   false, a, false, br, (short)0, racc, false, false);
    }

    const int col   = colbase + m;
    const float bi  = b[col];
    const float rbi = rb[col];

    #pragma unroll
    for (int v = 0; v < 8; ++v) {
        const int row = rowbase + v + 8 * ksel;
        const float hv = hacc[v];
        float rv = racc[v] + rbi;
        rv = rv > 0.0f ? rv : 0.0f;              // ReLU
        const float dn = dinv[row];
        h_out  [(size_t)row * DD + col] = hv;
        acc_out[(size_t)row * DD + col] = rv + bi + hv * dn * dn;
    }
}

// ---------------- edge scatter: acc[dst] += h[src] * dinv[src]*dinv[dst] ----
// One wave per edge; 4 coalesced 128B loads + 4 coalesced f32 atomics.
// h (51.2 MB) and acc (51.2 MB) both fit in the 192 MB L2, so this phase
// runs at L2 atomic bandwidth, not HBM.
__global__ __launch_bounds__(256) void edge_scatter(
    const int* __restrict__ ei, const float* __restrict__ dinv,
    const float* __restrict__ h, float* __restrict__ acc)
{
    const int e = blockIdx.x * 8 + (threadIdx.x >> 5);
    if (e >= EE) return;
    const int lane = threadIdx.x & 31;
    const int src = ei[e];
    const int dst = ei[EE + e];
    const float w = dinv[src] * dinv[dst];

    const float* __restrict__ hp = h + (size_t)src * DD;
    float* __restrict__ ap = acc + (size_t)dst * DD;
    #pragma unroll
    for (int j = 0; j < 4; ++j) {
        const int f = j * 32 + lane;
        unsafeAtomicAdd(&ap[f], hp[f] * w);
    }
}

// ---------------- BatchNorm: per-feature sum / sumsq ----------------
__global__ __launch_bounds__(128) void bn_stats(
    const float* __restrict__ acc, float* __restrict__ stats)
{
    const int f = threadIdx.x;                   // feature id
    float s = 0.0f, s2 = 0.0f;
    for (int r = blockIdx.x; r < NN; r += gridDim.x) {
        const float v = acc[(size_t)r * DD + f];
        s  += v;
        s2 += v * v;
    }
    unsafeAtomicAdd(&stats[f], s);
    unsafeAtomicAdd(&stats[DD + f], s2);
}

__global__ __launch_bounds__(256) void bn_apply(
    const float* __restrict__ acc, const float* __restrict__ stats,
    const float* __restrict__ gamma, const float* __restrict__ beta,
    float* __restrict__ xout)
{
    const int i = blockIdx.x * blockDim.x + threadIdx.x;
    if (i >= NN * DD) return;
    const int f = i & (DD - 1);
    const float inv_n = 1.0f / (float)NN;
    const float mean = stats[f] * inv_n;
    const float var  = stats[DD + f] * inv_n - mean * mean;   // biased
    const float isd  = __frsqrt_rn(var + BN_EPS);
    xout[i] = gamma[f] * (acc[i] - mean) * isd + beta[f];
}

// ---------------- global_add_pool ----------------
__global__ __launch_bounds__(128) void pool_add(
    const float* __restrict__ x, const int* __restrict__ batch,
    float* __restrict__ out)
{
    const int node = blockIdx.x;
    const int f = threadIdx.x;
    const int g = batch[node];
    unsafeAtomicAdd(&out[(size_t)g * DD + f], x[(size_t)node * DD + f]);
}

// ---------------- host launcher ----------------
extern "C" void kernel_launch(void* const* d_in, const int* in_sizes, int n_in,
                              void* d_out, int out_size, void* d_ws, size_t ws_size,
                              hipStream_t stream) {
    const float* x_in   = (const float*)d_in[0];
    const int*   ei     = (const int*)  d_in[1];
    const int*   batch  = (const int*)  d_in[2];
    const float* Ws     = (const float*)d_in[3];
    const float* bs     = (const float*)d_in[4];
    const float* rWs    = (const float*)d_in[5];
    const float* rbs    = (const float*)d_in[6];
    const float* gammas = (const float*)d_in[7];
    const float* betas  = (const float*)d_in[8];
    float* out = (float*)d_out;

    // workspace layout (floats)
    float* ws    = (float*)d_ws;
    float* hbuf  = ws;                            // N*D
    float* accb  = hbuf + (size_t)NN * DD;        // N*D
    float* xbuf  = accb + (size_t)NN * DD;        // N*D
    float* dinv  = xbuf + (size_t)NN * DD;        // N   (deg, then rsqrt in-place)
    float* stats = dinv + NN;                     // 2*D

    // degree with self-loops -> dinv
    fill_f32<<<(NN + 255) / 256, 256, 0, stream>>>(dinv, 1.0f, NN);
    deg_accum<<<(EE + 255) / 256, 256, 0, stream>>>(ei, dinv);
    deg_to_dinv<<<(NN + 255) / 256, 256, 0, stream>>>(dinv);

    // zero output accumulator
    fill_f32<<<(GG * DD + 255) / 256, 256, 0, stream>>>(out, 0.0f, GG * DD);

    const float* xcur = x_in;
    for (int l = 0; l < LL; ++l) {
        fill_f32<<<1, 256, 0, stream>>>(stats, 0.0f, 2 * DD);

        gemm_dual_wmma<<<NN / 16, 256, 0, stream>>>(
            xcur,
            Ws  + (size_t)l * DD * DD, rWs + (size_t)l * DD * DD,
            bs  + (size_t)l * DD,      rbs + (size_t)l * DD,
            dinv, hbuf, accb);

        edge_scatter<<<EE / 8, 256, 0, stream>>>(ei, dinv, hbuf, accb);

        bn_stats<<<512, 128, 0, stream>>>(accb, stats);

        bn_apply<<<(NN * DD + 255) / 256, 256, 0, stream>>>(
            accb, stats,
            gammas + (size_t)l * DD, betas + (size_t)l * DD, xbuf);

        xcur = xbuf;
    }

    pool_add<<<NN, 128, 0, stream>>>(xbuf, batch, out);
}